// Net_37958920962283
// MI455X (gfx1250) — compile-verified
//
#include <hip/hip_runtime.h>
#include <math.h>

typedef __attribute__((ext_vector_type(2))) float v2f;
typedef __attribute__((ext_vector_type(8))) float v8f;

#define NF 128   // input features
#define NH 16    // hidden

// ---------- small utility kernels ----------
__global__ void k_zero(float* __restrict__ p, int n) {
  int i = blockIdx.x * blockDim.x + threadIdx.x;
  if (i < n) p[i] = 0.0f;
}

// deg[c] += ew[e] for every edge (col destination, matching reference segment_sum on col)
__global__ void k_deg(const long long* __restrict__ col, const float* __restrict__ ew,
                      float* __restrict__ deg, int E) {
  int e = blockIdx.x * blockDim.x + threadIdx.x;
  if (e >= E) return;
  atomicAdd(&deg[(int)col[e]], ew[e]);
}

// dinv = rsqrt(deg + 1)   (self-loop adds weight 1; guard matches reference)
__global__ void k_dinv(float* __restrict__ deg, int n) {
  int i = blockIdx.x * blockDim.x + threadIdx.x;
  if (i >= n) return;
  float d = deg[i] + 1.0f;
  deg[i] = (d > 0.0f) ? rsqrtf(d) : 0.0f;
}

// ---------- h0 = x @ W1 via V_WMMA_F32_16X16X4_F32 ----------
// One wave per 16-row tile. A: 16x4 fp32 (lane half = K pair, v[0..1] = K within pair).
// B: 4x16 fp32 (N = lane%16, K pair by lane half, K-in-pair by VGPR).
// D: 8 VGPRs, M = r + 8*(lane>=16), N = lane%16.
__global__ void k_gemm1_wmma(const float* __restrict__ x, const float* __restrict__ W1,
                             float* __restrict__ h0, int n) {
  const int lane = threadIdx.x & 31;
  const int wave = blockIdx.x * (blockDim.x >> 5) + (threadIdx.x >> 5);
  const int row0 = wave * 16;
  if (row0 >= n) return;                    // wave-uniform exit: EXEC stays all-ones for WMMA
  const int m    = lane & 15;
  const int half = lane >> 4;

  const float* ap = x + (size_t)(row0 + m) * NF + 2 * half;
  v8f acc = {};
  #pragma unroll 4
  for (int k0 = 0; k0 < NF; k0 += 4) {
    v2f a, b;
    a.x = ap[k0];                           // x[row][k0 + 2*half]
    a.y = ap[k0 + 1];                       // x[row][k0 + 2*half + 1]
    const int kb = k0 + 2 * half;
    b.x = W1[kb * NH + m];                  // W1[kb][m]
    b.y = W1[(kb + 1) * NH + m];            // W1[kb+1][m]
    acc = __builtin_amdgcn_wmma_f32_16x16x4_f32(
        /*neg_a=*/false, a, /*neg_b=*/false, b,
        /*c_mod=*/(short)0, acc, /*reuse_a=*/false, /*reuse_b=*/false);
  }
  float* out = h0 + (size_t)row0 * NH + m;
  #pragma unroll
  for (int r = 0; r < 8; ++r)
    out[(size_t)(r + 8 * half) * NH] = acc[r];
}

// h1 = dinv^2 * h0 + b1   (self-loop contribution + bias, also zero-initializes h1)
__global__ void k_h1init(const float* __restrict__ h0, const float* __restrict__ dinv,
                         const float* __restrict__ b1, float* __restrict__ h1, int n) {
  int t = blockIdx.x * blockDim.x + threadIdx.x;
  if (t >= n * NH) return;
  int i = t >> 4, j = t & 15;
  float di = dinv[i];
  h1[t] = di * di * h0[t] + b1[j];
}

// per edge: h1[col] += dinv[row]*ew*dinv[col] * h0[row]; 16 lanes per edge (coalesced atomics)
__global__ void k_scatter1(const long long* __restrict__ row, const long long* __restrict__ col,
                           const float* __restrict__ ew, const float* __restrict__ dinv,
                           const float* __restrict__ h0, float* __restrict__ h1, int E) {
  int t = blockIdx.x * blockDim.x + threadIdx.x;
  int e = t >> 4;
  int j = t & 15;
  if (e >= E) return;
  int r = (int)row[e];
  int c = (int)col[e];
  float norm = dinv[r] * ew[e] * dinv[c];
  float v = h0[(size_t)r * NH + j] * norm;
  atomicAdd(&h1[(size_t)c * NH + j], v);
}

// z0 = relu(h1) @ W2   (16 -> 1)
__global__ void k_relu_gemm2(const float* __restrict__ h1, const float* __restrict__ W2,
                             float* __restrict__ z0, int n) {
  int i = blockIdx.x * blockDim.x + threadIdx.x;
  if (i >= n) return;
  const float* hp = h1 + (size_t)i * NH;
  float s = 0.0f;
  #pragma unroll
  for (int j = 0; j < NH; ++j) s += fmaxf(hp[j], 0.0f) * W2[j];
  z0[i] = s;
}

// out = dinv^2 * z0 + b2   (self-loop + bias; initializes d_out deterministically)
__global__ void k_outinit(const float* __restrict__ z0, const float* __restrict__ dinv,
                          const float* __restrict__ b2, float* __restrict__ out, int n) {
  int i = blockIdx.x * blockDim.x + threadIdx.x;
  if (i >= n) return;
  float di = dinv[i];
  out[i] = di * di * z0[i] + b2[0];
}

// per edge: out[col] += norm * z0[row]
__global__ void k_scatter2(const long long* __restrict__ row, const long long* __restrict__ col,
                           const float* __restrict__ ew, const float* __restrict__ dinv,
                           const float* __restrict__ z0, float* __restrict__ out, int E) {
  int e = blockIdx.x * blockDim.x + threadIdx.x;
  if (e >= E) return;
  int r = (int)row[e];
  int c = (int)col[e];
  float norm = dinv[r] * ew[e] * dinv[c];
  atomicAdd(&out[c], norm * z0[r]);
}

__global__ void k_sigmoid(float* __restrict__ out, int n) {
  int i = blockIdx.x * blockDim.x + threadIdx.x;
  if (i >= n) return;
  out[i] = 1.0f / (1.0f + expf(-out[i]));
}

// ---------- launch ----------
extern "C" void kernel_launch(void* const* d_in, const int* in_sizes, int n_in,
                              void* d_out, int out_size, void* d_ws, size_t ws_size,
                              hipStream_t stream) {
  const float*     x   = (const float*)d_in[0];
  const long long* ei  = (const long long*)d_in[1];   // int64 edge_index [2, E]
  const float*     ew  = (const float*)d_in[2];
  const float*     W1  = (const float*)d_in[3];
  const float*     b1  = (const float*)d_in[4];
  const float*     W2  = (const float*)d_in[5];
  const float*     b2  = (const float*)d_in[6];
  float*           out = (float*)d_out;

  const int n = in_sizes[0] / NF;
  const int E = in_sizes[1] / 2;
  const long long* row = ei;
  const long long* col = ei + E;

  // workspace layout (floats): dinv[n] | h0[n*16] | h1[n*16] | z0[n]
  float* ws   = (float*)d_ws;
  float* dinv = ws;
  float* h0   = dinv + n;
  float* h1   = h0 + (size_t)n * NH;
  float* z0   = h1 + (size_t)n * NH;

  const int B = 256;
  // degree -> dinv
  k_zero<<<(n + B - 1) / B, B, 0, stream>>>(dinv, n);
  k_deg<<<(E + B - 1) / B, B, 0, stream>>>(col, ew, dinv, E);
  k_dinv<<<(n + B - 1) / B, B, 0, stream>>>(dinv, n);
  // layer 1
  const int tiles = (n + 15) / 16;
  const int wpb = B / 32;  // waves per block
  k_gemm1_wmma<<<(tiles + wpb - 1) / wpb, B, 0, stream>>>(x, W1, h0, n);
  k_h1init<<<((long long)n * NH + B - 1) / B, B, 0, stream>>>(h0, dinv, b1, h1, n);
  k_scatter1<<<((long long)E * NH + B - 1) / B, B, 0, stream>>>(row, col, ew, dinv, h0, h1, E);
  // layer 2
  k_relu_gemm2<<<(n + B - 1) / B, B, 0, stream>>>(h1, W2, z0, n);
  k_outinit<<<(n + B - 1) / B, B, 0, stream>>>(z0, dinv, b2, out, n);
  k_scatter2<<<(E + B - 1) / B, B, 0, stream>>>(row, col, ew, dinv, z0, out, E);
  k_sigmoid<<<(n + B - 1) / B, B, 0, stream>>>(out, n);
}